// VanillaGNNLayer_32890859553163
// MI455X (gfx1250) — compile-verified
//
#include <hip/hip_runtime.h>

#define N_NODES 12288
#define DIN     64
#define DOUT    32
#define N_EDGES 393216

typedef __attribute__((ext_vector_type(2))) float v2f;
typedef __attribute__((ext_vector_type(8))) float v8f;

// ---------------------------------------------------------------------------
// 1) zero deg[] and acc[]
// ---------------------------------------------------------------------------
__global__ void gnn_zero_kernel(int* __restrict__ deg, float* __restrict__ acc,
                                int n_deg, int n_acc) {
    int i = blockIdx.x * blockDim.x + threadIdx.x;
    if (i < n_deg) deg[i] = 0;
    if (i < n_acc) acc[i] = 0.0f;
}

// ---------------------------------------------------------------------------
// 2) deg[j] = #edges with edge_index[1]==j   (column sums of A)
// ---------------------------------------------------------------------------
__global__ void gnn_degree_kernel(const int* __restrict__ edst, int* __restrict__ deg) {
    int e = blockIdx.x * blockDim.x + threadIdx.x;
    if (e < N_EDGES) atomicAdd(&deg[edst[e]], 1);
}

// ---------------------------------------------------------------------------
// 3) d[j] = rsqrt(1 + deg[j])
// ---------------------------------------------------------------------------
__global__ void gnn_dvec_kernel(const int* __restrict__ deg, float* __restrict__ dvec) {
    int i = blockIdx.x * blockDim.x + threadIdx.x;
    if (i < N_NODES) dvec[i] = rsqrtf(1.0f + (float)deg[i]);
}

// ---------------------------------------------------------------------------
// 4) g = d ⊙ (x @ W^T) via V_WMMA_F32_16X16X4_F32.
//    One wave per 16x16 output tile; 16 WMMA steps cover K=64.
//    A(16x4) lane(m=lane&15, kh=lane>>4): VGPR0/1 = X[m][2kh], X[m][2kh+1]
//    B(4x16) lane(n=lane&15, kh):         VGPR0/1 = W[n][2kh], W[n][2kh+1]  (B=W^T)
//    C/D:    VGPR v holds D[v + 8*kh][lane&15]
// ---------------------------------------------------------------------------
__global__ __launch_bounds__(128) void
gnn_gemm_wmma_kernel(const float* __restrict__ x, const float* __restrict__ W,
                     const float* __restrict__ dvec, float* __restrict__ g) {
    const int lane  = threadIdx.x & 31;
    const int wave  = threadIdx.x >> 5;
    const int tile  = blockIdx.x * 4 + wave;   // 1536 tiles total (768 x 2)
    const int tm    = tile >> 1;
    const int tn    = tile & 1;
    const int row0  = tm * 16;
    const int col0  = tn * 16;

    const int m  = lane & 15;
    const int kh = lane >> 4;                  // 0 or 1

    const float* xrow = x + (size_t)(row0 + m) * DIN + 2 * kh;
    const float* wrow = W + (size_t)(col0 + m) * DIN + 2 * kh;

    v8f c = {};
#pragma unroll
    for (int k0 = 0; k0 < DIN; k0 += 4) {
        v2f a = *(const v2f*)(xrow + k0);
        v2f b = *(const v2f*)(wrow + k0);
        c = __builtin_amdgcn_wmma_f32_16x16x4_f32(
                /*neg_a=*/false, a, /*neg_b=*/false, b,
                /*c_mod=*/(short)0, c, /*reuse_a=*/false, /*reuse_b=*/false);
    }

    const int col = col0 + (lane & 15);
#pragma unroll
    for (int v = 0; v < 8; ++v) {
        const int r = row0 + v + 8 * kh;
        g[(size_t)r * DOUT + col] = c[v] * dvec[r];
    }
}

// ---------------------------------------------------------------------------
// 5) for each edge e: acc[dst[e]] += g[src[e]]   (one wave per edge; lane = channel)
// ---------------------------------------------------------------------------
__global__ void gnn_scatter_kernel(const int* __restrict__ esrc, const int* __restrict__ edst,
                                   const float* __restrict__ g, float* __restrict__ acc) {
    const long long tid = (long long)blockIdx.x * blockDim.x + threadIdx.x;
    const int e  = (int)(tid >> 5);   // DOUT == 32 == wave32 width
    const int ch = (int)(tid & 31);
    if (e < N_EDGES) {
        const int s = esrc[e];
        const int t = edst[e];
        const float val = g[(size_t)s * DOUT + ch];
        atomicAdd(&acc[(size_t)t * DOUT + ch], val);
    }
}

// ---------------------------------------------------------------------------
// 6) out[i] = d[i] * (g[i] + acc[i])     (self-loop term folded in: d*g = d^2*h)
// ---------------------------------------------------------------------------
__global__ void gnn_finalize_kernel(const float* __restrict__ g, const float* __restrict__ acc,
                                    const float* __restrict__ dvec, float* __restrict__ out) {
    int i = blockIdx.x * blockDim.x + threadIdx.x;
    if (i < N_NODES * DOUT) {
        int node = i >> 5;            // DOUT == 32
        out[i] = dvec[node] * (g[i] + acc[i]);
    }
}

extern "C" void kernel_launch(void* const* d_in, const int* in_sizes, int n_in,
                              void* d_out, int out_size, void* d_ws, size_t ws_size,
                              hipStream_t stream) {
    const float* x  = (const float*)d_in[0];   // [N, DIN]
    const float* W  = (const float*)d_in[1];   // [DOUT, DIN]
    const int*   ei = (const int*)d_in[2];     // [2, E] flat
    const int* esrc = ei;                      // edge_index[0]
    const int* edst = ei + N_EDGES;            // edge_index[1]
    float* out = (float*)d_out;

    char* ws   = (char*)d_ws;
    int*   deg  = (int*)ws;                                       // N ints
    float* dvec = (float*)(ws + (size_t)N_NODES * 4);             // N floats
    float* g    = (float*)(ws + (size_t)N_NODES * 8);             // N*32 floats
    float* acc  = g + (size_t)N_NODES * DOUT;                     // N*32 floats

    gnn_zero_kernel<<<(N_NODES * DOUT + 255) / 256, 256, 0, stream>>>(
        deg, acc, N_NODES, N_NODES * DOUT);
    gnn_degree_kernel<<<(N_EDGES + 255) / 256, 256, 0, stream>>>(edst, deg);
    gnn_dvec_kernel<<<(N_NODES + 255) / 256, 256, 0, stream>>>(deg, dvec);
    gnn_gemm_wmma_kernel<<<384, 128, 0, stream>>>(x, W, dvec, g);
    gnn_scatter_kernel<<<(N_EDGES * 32) / 256, 256, 0, stream>>>(esrc, edst, g, acc);
    gnn_finalize_kernel<<<(N_NODES * DOUT + 255) / 256, 256, 0, stream>>>(g, acc, dvec, out);
}